// GraphDenoisingModule_88313117540794
// MI455X (gfx1250) — compile-verified
//
#include <hip/hip_runtime.h>
#include <math.h>

typedef __attribute__((ext_vector_type(2))) float v2f;
typedef __attribute__((ext_vector_type(8))) float v8f;

#define NEG_SLOPE 0.2f
#define EPSV 1e-16f

// ---------- order-preserving f32 <-> u32 encoding (for atomicMax on floats) ----------
__device__ __forceinline__ unsigned ford(float f) {
  unsigned u = __float_as_uint(f);
  return (u & 0x80000000u) ? ~u : (u | 0x80000000u);
}
__device__ __forceinline__ float funord(unsigned u) {
  return __uint_as_float((u & 0x80000000u) ? (u & 0x7FFFFFFFu) : ~u);
}
// encoding of -inf: ~0xFF800000 = 0x007FFFFF
#define NEG_INF_ENC 0x007FFFFFu

// ---------- dense GEMM: C[M,Nc] = A[M,K] @ B[K,Nc], fp32 WMMA 16x16x4 ----------
// One wave computes a 64x16 slab (4 M-tiles sharing one N-tile). Per k-step:
// load B fragment once + 4 A fragments into distinct regs, then 4 independent
// v_wmma_f32_16x16x4_f32 back-to-back (loads clause together, single wait).
// Wave32; EXEC stays all-1s around the WMMAs (loads clamped, stores guarded).
// A lane layout (16x4 f32): lanes 0-15 -> M=lane, K={0,1}; lanes 16-31 -> M=lane-16, K={2,3}
// B lane layout (4x16 f32): v0: K=0 (lanes0-15)/K=2 (lanes16-31); v1: K=1/K=3, N=lane&15
// C/D layout (16x16 f32):  vgpr r, lanes0-15 -> M=r, lanes16-31 -> M=8+r, N=lane&15
__global__ void gemm_f32_wmma(const float* __restrict__ A,
                              const float* __restrict__ B,
                              float* __restrict__ C,
                              int M, int K, int Nc) {
  const int lane   = threadIdx.x & 31;
  const int wid    = threadIdx.x >> 5;
  const int tilesN = Nc >> 4;
  const int slabsM = (M + 63) >> 6;               // 64-row slabs
  const int slab   = blockIdx.x * (blockDim.x >> 5) + wid;
  if (slab >= slabsM * tilesN) return;
  const int sm = slab / tilesN;                    // slab row index
  const int tn = slab - sm * tilesN;               // N tile

  const int ml = lane & 15;
  const int hi = lane >> 4;
  const int m0 = sm * 64;
  const int bcol = tn * 16 + ml;

  // clamped A row pointers for the 4 tiles (keeps wave uniform, loads in range)
  int r0 = m0 +  0 + ml; if (r0 >= M) r0 = M - 1;
  int r1 = m0 + 16 + ml; if (r1 >= M) r1 = M - 1;
  int r2 = m0 + 32 + ml; if (r2 >= M) r2 = M - 1;
  int r3 = m0 + 48 + ml; if (r3 >= M) r3 = M - 1;
  const float* ap0 = A + (size_t)r0 * K + 2 * hi;
  const float* ap1 = A + (size_t)r1 * K + 2 * hi;
  const float* ap2 = A + (size_t)r2 * K + 2 * hi;
  const float* ap3 = A + (size_t)r3 * K + 2 * hi;

  const float* bp = B + (size_t)(2 * hi) * Nc + bcol;
  const size_t bstep = (size_t)4 * Nc;

  v8f acc0 = {}, acc1 = {}, acc2 = {}, acc3 = {};
  for (int k0 = 0; k0 < K; k0 += 4) {
    // ---- all loads first (distinct registers -> one clause, one wait) ----
    const float2 f0 = *reinterpret_cast<const float2*>(ap0);
    const float2 f1 = *reinterpret_cast<const float2*>(ap1);
    const float2 f2 = *reinterpret_cast<const float2*>(ap2);
    const float2 f3 = *reinterpret_cast<const float2*>(ap3);
    const float  bl = bp[0];
    const float  bh = bp[Nc];
    ap0 += 4; ap1 += 4; ap2 += 4; ap3 += 4; bp += bstep;

    v2f b;  b[0] = bl;   b[1] = bh;
    v2f a0; a0[0] = f0.x; a0[1] = f0.y;
    v2f a1; a1[0] = f1.x; a1[1] = f1.y;
    v2f a2; a2[0] = f2.x; a2[1] = f2.y;
    v2f a3; a3[0] = f3.x; a3[1] = f3.y;

    // ---- 4 independent WMMAs back-to-back ----
    acc0 = __builtin_amdgcn_wmma_f32_16x16x4_f32(false, a0, false, b, (short)0, acc0, false, false);
    acc1 = __builtin_amdgcn_wmma_f32_16x16x4_f32(false, a1, false, b, (short)0, acc1, false, false);
    acc2 = __builtin_amdgcn_wmma_f32_16x16x4_f32(false, a2, false, b, (short)0, acc2, false, false);
    acc3 = __builtin_amdgcn_wmma_f32_16x16x4_f32(false, a3, false, b, (short)0, acc3, false, false);
  }

  // store 4 tiles
  const int mrel = 8 * hi;
#pragma unroll
  for (int r = 0; r < 8; ++r) {
    int row;
    row = m0 +  0 + mrel + r; if (row < M) C[(size_t)row * Nc + bcol] = acc0[r];
    row = m0 + 16 + mrel + r; if (row < M) C[(size_t)row * Nc + bcol] = acc1[r];
    row = m0 + 32 + mrel + r; if (row < M) C[(size_t)row * Nc + bcol] = acc2[r];
    row = m0 + 48 + mrel + r; if (row < M) C[(size_t)row * Nc + bcol] = acc3[r];
  }
}

// ---------- per-node attention logits: aS[n] = h[n,:].a_src, aD[n] = h[n,:].a_dst ----------
__global__ void node_alpha_k(const float* __restrict__ h,
                             const float* __restrict__ vs,
                             const float* __restrict__ vd,
                             float* __restrict__ oS, float* __restrict__ oD,
                             int N, int F) {
  const int gid  = blockIdx.x * blockDim.x + threadIdx.x;
  const int n    = gid >> 5;
  const int lane = gid & 31;
  if (n >= N) return;
  const float* hp = h + (size_t)n * F;
  float ss = 0.f, sd = 0.f;
  for (int f = lane; f < F; f += 32) {
    const float v = hp[f];
    ss = fmaf(v, vs[f], ss);
    sd = fmaf(v, vd[f], sd);
  }
  for (int off = 16; off > 0; off >>= 1) {
    ss += __shfl_down(ss, off);
    sd += __shfl_down(sd, off);
  }
  if (lane == 0) { oS[n] = ss; oD[n] = sd; }
}

// ---------- small elementwise helpers ----------
__global__ void init_softmax_k(unsigned* __restrict__ m, float* __restrict__ s, int n) {
  const int i = blockIdx.x * blockDim.x + threadIdx.x;
  if (i < n) { m[i] = NEG_INF_ENC; s[i] = 0.f; }
}
__global__ void zero_f32_k(float* __restrict__ p, int n) {
  const int i = blockIdx.x * blockDim.x + threadIdx.x;
  if (i < n) p[i] = 0.f;
}
__global__ void bias_relu_k(float* __restrict__ p, const float* __restrict__ b,
                            int total, int F) {
  const int i = blockIdx.x * blockDim.x + threadIdx.x;
  if (i < total) {
    const float v = p[i] + b[i & (F - 1)];   // F is a power of two (64/128)
    p[i] = v > 0.f ? v : 0.f;
  }
}
__global__ void bias_init_k(float* __restrict__ p, const float* __restrict__ b,
                            int total, int F) {
  const int i = blockIdx.x * blockDim.x + threadIdx.x;
  if (i < total) p[i] = b[i & (F - 1)];
}

// ---------- edge helpers ----------
__device__ __forceinline__ void edge_sd(const int* __restrict__ ei, int E, int e,
                                        int& s, int& d) {
  if (e < E) { s = ei[e]; d = ei[E + e]; }   // edge_index rows [src; dst]
  else       { s = d = e - E; }               // appended self-loops
}
__device__ __forceinline__ float edge_logit(const float* __restrict__ aS,
                                            const float* __restrict__ aD,
                                            int s, int d) {
  const float v = aS[s] + aD[d];
  return v > 0.f ? v : NEG_SLOPE * v;        // leaky_relu
}

__global__ void edge_max_k(const int* __restrict__ ei, int E, int Et,
                           const float* __restrict__ aS, const float* __restrict__ aD,
                           unsigned* __restrict__ menc) {
  const int e = blockIdx.x * blockDim.x + threadIdx.x;
  if (e >= Et) return;
  int s, d; edge_sd(ei, E, e, s, d);
  atomicMax(&menc[d], ford(edge_logit(aS, aD, s, d)));
}

__global__ void edge_exp_k(const int* __restrict__ ei, int E, int Et,
                           const float* __restrict__ aS, const float* __restrict__ aD,
                           const unsigned* __restrict__ menc,
                           float* __restrict__ ssum, float* __restrict__ exb) {
  const int e = blockIdx.x * blockDim.x + threadIdx.x;
  if (e >= Et) return;
  int s, d; edge_sd(ei, E, e, s, d);
  const float ex = expf(edge_logit(aS, aD, s, d) - funord(menc[d]));
  exb[e] = ex;
  atomicAdd(&ssum[d], ex);
}

// one wave per edge: out[dst,:] += alpha * h[src,:]
__global__ void edge_scatter_k(const int* __restrict__ ei, int E, int Et,
                               const float* __restrict__ exb,
                               const float* __restrict__ ssum,
                               const float* __restrict__ h,
                               float* __restrict__ out, int F) {
  const int gid  = blockIdx.x * blockDim.x + threadIdx.x;
  const int e    = gid >> 5;
  const int lane = gid & 31;
  if (e >= Et) return;
  int s, d; edge_sd(ei, E, e, s, d);
  const float alpha = exb[e] / (ssum[d] + EPSV);
  const float* __restrict__ hs = h + (size_t)s * F;
  float* __restrict__ od = out + (size_t)d * F;
  for (int f = lane; f < F; f += 32)
    atomicAdd(&od[f], alpha * hs[f]);
}

// ---------------------------------------------------------------------------
extern "C" void kernel_launch(void* const* d_in, const int* in_sizes, int n_in,
                              void* d_out, int out_size, void* d_ws, size_t ws_size,
                              hipStream_t stream) {
  const float* x   = (const float*)d_in[0];
  const int*   ei  = (const int*)  d_in[1];
  const float* W1  = (const float*)d_in[2];
  const float* as1 = (const float*)d_in[3];
  const float* ad1 = (const float*)d_in[4];
  const float* b1  = (const float*)d_in[5];
  const float* W2  = (const float*)d_in[6];
  const float* as2 = (const float*)d_in[7];
  const float* ad2 = (const float*)d_in[8];
  const float* b2  = (const float*)d_in[9];

  const int HID  = in_sizes[3];            // 64
  const int COUT = in_sizes[7];            // 128
  const int CIN  = in_sizes[2] / HID;      // 128
  const int N    = in_sizes[0] / CIN;      // 50000
  const int E    = in_sizes[1] / 2;        // 1.2M
  const int Et   = E + N;                  // with self-loops

  // workspace carve-out (256B aligned regions)
  char* ws = (char*)d_ws;
  auto carve = [&](size_t bytes) -> char* {
    char* p = ws; ws += (bytes + 255) & ~(size_t)255; return p;
  };
  float*    h1   = (float*)   carve((size_t)N * HID  * 4);
  float*    acc1 = (float*)   carve((size_t)N * HID  * 4);
  float*    h2   = (float*)   carve((size_t)N * COUT * 4);
  float*    aS   = (float*)   carve((size_t)N * 4);
  float*    aD   = (float*)   carve((size_t)N * 4);
  unsigned* menc = (unsigned*)carve((size_t)N * 4);
  float*    ssum = (float*)   carve((size_t)N * 4);
  float*    exb  = (float*)   carve((size_t)Et * 4);
  (void)ws_size; (void)n_in; (void)out_size;

  float* out = (float*)d_out;

  const int TB = 256;
  auto cdiv = [](int a, int b) { return (a + b - 1) / b; };

  // =========================== layer 1 ===========================
  {
    const int slabs = cdiv(N, 64) * (HID >> 4);    // 64x16 slabs, 4 waves/block
    gemm_f32_wmma<<<cdiv(slabs, 4), 128, 0, stream>>>(x, W1, h1, N, CIN, HID);
  }
  node_alpha_k<<<cdiv(N * 32, TB), TB, 0, stream>>>(h1, as1, ad1, aS, aD, N, HID);
  init_softmax_k<<<cdiv(N, TB), TB, 0, stream>>>(menc, ssum, N);
  zero_f32_k<<<cdiv(N * HID, TB), TB, 0, stream>>>(acc1, N * HID);
  edge_max_k<<<cdiv(Et, TB), TB, 0, stream>>>(ei, E, Et, aS, aD, menc);
  edge_exp_k<<<cdiv(Et, TB), TB, 0, stream>>>(ei, E, Et, aS, aD, menc, ssum, exb);
  edge_scatter_k<<<cdiv(Et, 8), TB, 0, stream>>>(ei, E, Et, exb, ssum, h1, acc1, HID);
  bias_relu_k<<<cdiv(N * HID, TB), TB, 0, stream>>>(acc1, b1, N * HID, HID);

  // =========================== layer 2 ===========================
  {
    const int slabs = cdiv(N, 64) * (COUT >> 4);
    gemm_f32_wmma<<<cdiv(slabs, 4), 128, 0, stream>>>(acc1, W2, h2, N, HID, COUT);
  }
  node_alpha_k<<<cdiv(N * 32, TB), TB, 0, stream>>>(h2, as2, ad2, aS, aD, N, COUT);
  init_softmax_k<<<cdiv(N, TB), TB, 0, stream>>>(menc, ssum, N);
  edge_max_k<<<cdiv(Et, TB), TB, 0, stream>>>(ei, E, Et, aS, aD, menc);
  edge_exp_k<<<cdiv(Et, TB), TB, 0, stream>>>(ei, E, Et, aS, aD, menc, ssum, exb);
  bias_init_k<<<cdiv(N * COUT, TB), TB, 0, stream>>>(out, b2, N * COUT, COUT);
  edge_scatter_k<<<cdiv(Et, 8), TB, 0, stream>>>(ei, E, Et, exb, ssum, h2, out, COUT);
}